// FNO2d_2353642078942
// MI455X (gfx1250) — compile-verified
//
#include <hip/hip_runtime.h>
#include <math.h>

// ---------------------------------------------------------------------------
// FNO2d forward for MI455X (gfx1250), fp32 via V_WMMA_F32_16X16X4_F32.
// Truncated-mode DFTs expressed as dense GEMMs against precomputed trig
// matrices; all big GEMMs run through one generic wave32 WMMA kernel.
// ---------------------------------------------------------------------------

typedef float v2f __attribute__((ext_vector_type(2)));
typedef float v8f __attribute__((ext_vector_type(8)));

#define BB 4
#define CC 3
#define HH 256
#define WW 256
#define LL 4
#define WIDTH 32
#define M1 12
#define M2 12
#define HWSZ (HH * WW)                  // 65536
#define HBUF (BB * WIDTH * HWSZ)        // 8388608 floats

__device__ __forceinline__ float gelu_exact(float v) {
    return 0.5f * v * (1.0f + erff(v * 0.70710678118654752440f));
}

// ---------------------------------------------------------------------------
// Generic fp32 WMMA GEMM:  C = A @ B  (+bias / +Add / gelu per `epi`)
//   A(m,k) = A[bz*Abat + m*As_m + k*As_k]
//   B(k,n) = Bm[k*Bs_k + n]                  (shared across batches)
//   C(m,n) = C[bz*Cbat + m*Cs_m + n*Cs_n]    (Add uses same addressing)
// One wave computes a 16 x (16*NT) tile. K must be a multiple of 4,
// M a multiple of 16, N (padded) a multiple of 16*NT. Store guarded by Nvalid.
// epi: 0 none, 1 +bias, 2 +bias+gelu, 3 +Add, 4 +Add+gelu
// ---------------------------------------------------------------------------
template <int NT>
__global__ __launch_bounds__(128) void gemm_wmma_k(
    const float* __restrict__ A, const float* __restrict__ Bm,
    const float* __restrict__ bias, const float* __restrict__ Add,
    float* __restrict__ C,
    int Mtiles, int NGroups, int K, int Nvalid, int epi,
    long long As_m, long long As_k, long long Abat,
    long long Bs_k,
    long long Cs_m, long long Cs_n, long long Cbat)
{
    int wid = blockIdx.x * blockDim.y + threadIdx.y;
    if (wid >= Mtiles * NGroups) return;           // wave-uniform: EXEC stays all-1s
    int mt = wid / NGroups;
    int ng = wid % NGroups;

    long long bz = blockIdx.z;
    const float* Ab = A + bz * Abat;
    float* Cb = C + bz * Cbat;
    const float* Addb = Add ? (Add + bz * Cbat) : nullptr;

    int lane = threadIdx.x;
    int l16 = lane & 15;
    int lh  = lane >> 4;                           // 0: lanes 0-15, 1: lanes 16-31
    int m0 = mt * 16;
    int n0 = ng * 16 * NT;

    v8f acc[NT] = {};

    const float* Arow = Ab + (long long)(m0 + l16) * As_m;
    for (int k0 = 0; k0 < K; k0 += 4) {
        int ka = k0 + 2 * lh;
        v2f va;
        va.x = Arow[(long long)ka * As_k];
        va.y = Arow[(long long)(ka + 1) * As_k];
        const float* Bp0 = Bm + (long long)ka * Bs_k + n0 + l16;
        const float* Bp1 = Bp0 + Bs_k;
#pragma unroll
        for (int t = 0; t < NT; ++t) {
            v2f vb;
            vb.x = Bp0[t * 16];
            vb.y = Bp1[t * 16];
            acc[t] = __builtin_amdgcn_wmma_f32_16x16x4_f32(
                false, va, false, vb, (short)0, acc[t], false, false);
        }
    }

#pragma unroll
    for (int t = 0; t < NT; ++t) {
        int n = n0 + t * 16 + l16;
        if (n >= Nvalid) continue;
        float bv = (epi == 1 || epi == 2) ? bias[n] : 0.0f;
#pragma unroll
        for (int r = 0; r < 8; ++r) {
            int m = m0 + r + 8 * lh;
            long long off = (long long)m * Cs_m + (long long)n * Cs_n;
            float v = acc[t][r] + bv;
            if (epi == 3 || epi == 4) v += Addb[off];
            if (epi == 2 || epi == 4) v = gelu_exact(v);
            Cb[off] = v;
        }
    }
}

// ---------------------------------------------------------------------------
// Coefficient tables (rebuilt deterministically every call).
//  E2f  [256][32] : y-DFT columns   (cos | -sin | zero-pad)
//  E1f  [24][256] : e^{+i 2pi kx x/H} pairs (cos,sin), kx in {0..11, -12..-1}
//  EI1  [24][256] : same / H  (inverse kx transform)
//  Cinv [24][256] : irfft ky rows:  wgt*cos/W  |  -wgt*sin/W
//  W1p  [128][16] : mlp1_w zero-padded to 16 cols; B1p[16] padded bias
// ---------------------------------------------------------------------------
__global__ void init_k(float* __restrict__ E2f, float2* __restrict__ E1f,
                       float2* __restrict__ EI1, float* __restrict__ Cinv,
                       float* __restrict__ W1p, float* __restrict__ B1p,
                       const float* __restrict__ mlp1_w,
                       const float* __restrict__ mlp1_b)
{
    int i = blockIdx.x * blockDim.x + threadIdx.x;
    const float TP = 6.28318530717958647692f;
    if (i < 256 * 32) {
        int y = i >> 5, c = i & 31;
        float v = 0.0f;
        if (c < 12)      v =  cosf(TP * (float)(c * y) * (1.0f / 256.0f));
        else if (c < 24) v = -sinf(TP * (float)((c - 12) * y) * (1.0f / 256.0f));
        E2f[i] = v;
        return;
    }
    i -= 256 * 32;
    if (i < 24 * 256) {
        int kxi = i >> 8, xx = i & 255;
        int kk = (kxi < 12) ? kxi : (kxi - 24);
        float ang = TP * (float)(kk * xx) * (1.0f / 256.0f);
        E1f[i] = make_float2(cosf(ang), sinf(ang));
        return;
    }
    i -= 24 * 256;
    if (i < 24 * 256) {
        int kxi = i >> 8, xx = i & 255;
        int kk = (kxi < 12) ? kxi : (kxi - 24);
        float ang = TP * (float)(kk * xx) * (1.0f / 256.0f);
        EI1[i] = make_float2(cosf(ang) * (1.0f / 256.0f), sinf(ang) * (1.0f / 256.0f));
        return;
    }
    i -= 24 * 256;
    if (i < 24 * 256) {
        int k = i >> 8, y = i & 255;
        int ky = (k < 12) ? k : (k - 12);
        float w = ((ky == 0) ? 1.0f : 2.0f) * (1.0f / 256.0f);
        float ang = TP * (float)(ky * y) * (1.0f / 256.0f);
        Cinv[i] = (k < 12) ? (w * cosf(ang)) : (-w * sinf(ang));
        return;
    }
    i -= 24 * 256;
    if (i < 128 * 16) {
        int r = i >> 4, c = i & 15;
        W1p[i] = (c < 3) ? mlp1_w[r * 3 + c] : 0.0f;
        return;
    }
    i -= 128 * 16;
    if (i < 16) {
        B1p[i] = (i < 3) ? mlp1_b[i] : 0.0f;
        return;
    }
}

// Lift: h[b,o,x,y] = p_b[o] + sum_c feat[c]*p_w[c,o], feat = (x0,x1,x2, x/255, y/255)
__global__ void lift_k(const float* __restrict__ xin, const float* __restrict__ p_w,
                       const float* __restrict__ p_b, float* __restrict__ h)
{
    int idx = blockIdx.x * blockDim.x + threadIdx.x;
    if (idx >= HBUF) return;
    int y  = idx & 255;
    int xr = (idx >> 8) & 255;
    int o  = (idx >> 16) & 31;
    int b  = idx >> 21;
    long long xb = (long long)b * (CC * HWSZ) + xr * 256 + y;
    float f0 = xin[xb];
    float f1 = xin[xb + HWSZ];
    float f2 = xin[xb + 2 * HWSZ];
    float gx = (float)xr * (1.0f / 255.0f);
    float gy = (float)y  * (1.0f / 255.0f);
    h[idx] = p_b[o] + f0 * p_w[o] + f1 * p_w[32 + o] + f2 * p_w[64 + o]
                    + gx * p_w[96 + o] + gy * p_w[128 + o];
}

// Stage B: HF[bc,kxi,ky] = sum_x T[bc,x,(ky|12+ky)] * e^{-i 2pi kx x/H}
__global__ void stageB_k(const float* __restrict__ T, const float2* __restrict__ E1f,
                         float2* __restrict__ HF)
{
    int idx = blockIdx.x * blockDim.x + threadIdx.x;
    if (idx >= BB * WIDTH * 24 * M2) return;
    int ky  = idx % 12;
    int kxi = (idx / 12) % 24;
    int bc  = idx / 288;
    const float* Tp = T + (long long)bc * 256 * 32;
    const float2* Ep = E1f + kxi * 256;
    float hr = 0.0f, hi = 0.0f;
    for (int x = 0; x < 256; ++x) {
        float tr = Tp[x * 32 + ky];
        float ti = Tp[x * 32 + 12 + ky];
        float2 e = Ep[x];                 // (cos, sin) of +2pi kx x / H
        hr += tr * e.x + ti * e.y;        // (tr + i ti) * (c - i s)
        hi += ti * e.x - tr * e.y;
    }
    HF[idx] = make_float2(hr, hi);
}

// Mode mixing: OUT[b,o,kxi,ky] = sum_i HF[b,i,kxi,ky] * W[l,i,o,m1,ky]
__global__ void mix_k(const float2* __restrict__ HF,
                      const float* __restrict__ sw1r, const float* __restrict__ sw1i,
                      const float* __restrict__ sw2r, const float* __restrict__ sw2i,
                      float2* __restrict__ OUTF, int l)
{
    int idx = blockIdx.x * blockDim.x + threadIdx.x;
    if (idx >= BB * WIDTH * 24 * M2) return;
    int ky  = idx % 12;
    int kxi = (idx / 12) % 24;
    int o   = (idx / 288) % 32;
    int b   = idx / (288 * 32);
    bool low = (kxi < 12);
    int m1 = low ? kxi : (kxi - 12);
    const float* wrb = low ? sw1r : sw2r;
    const float* wib = low ? sw1i : sw2i;
    float orr = 0.0f, oii = 0.0f;
    for (int i = 0; i < 32; ++i) {
        long long widx = ((((long long)l * 32 + i) * 32 + o) * 144) + m1 * 12 + ky;
        float wr = wrb[widx], wi = wib[widx];
        float2 hv = HF[((b * 32 + i) * 24 + kxi) * 12 + ky];
        orr += hv.x * wr - hv.y * wi;
        oii += hv.x * wi + hv.y * wr;
    }
    OUTF[((b * 32 + o) * 24 + kxi) * 12 + ky] = make_float2(orr, oii);
}

// Stage C: G[bo,x, ky | 12+ky] = sum_kxi OUT[bo,kxi,ky] * e^{+i 2pi kx x/H} / H
__global__ void stageC_k(const float2* __restrict__ OUTF, const float2* __restrict__ EI1,
                         float* __restrict__ G)
{
    int idx = blockIdx.x * blockDim.x + threadIdx.x;
    if (idx >= BB * WIDTH * 256) return;
    int x  = idx % 256;
    int bo = idx / 256;
    const float2* Op = OUTF + (long long)bo * 288;
    float gr[12], gi[12];
#pragma unroll
    for (int k = 0; k < 12; ++k) { gr[k] = 0.0f; gi[k] = 0.0f; }
    for (int kxi = 0; kxi < 24; ++kxi) {
        float2 e = EI1[kxi * 256 + x];
#pragma unroll
        for (int ky = 0; ky < 12; ++ky) {
            float2 ov = Op[kxi * 12 + ky];
            gr[ky] += ov.x * e.x - ov.y * e.y;
            gi[ky] += ov.x * e.y + ov.y * e.x;
        }
    }
    float* Gp = G + (long long)idx * 24;
#pragma unroll
    for (int ky = 0; ky < 12; ++ky) { Gp[ky] = gr[ky]; Gp[12 + ky] = gi[ky]; }
}

// ---------------------------------------------------------------------------
extern "C" void kernel_launch(void* const* d_in, const int* in_sizes, int n_in,
                              void* d_out, int out_size, void* d_ws, size_t ws_size,
                              hipStream_t stream)
{
    (void)in_sizes; (void)n_in; (void)out_size;
    const float* xin    = (const float*)d_in[0];
    const float* p_w    = (const float*)d_in[1];
    const float* p_b    = (const float*)d_in[2];
    const float* sw1r   = (const float*)d_in[3];
    const float* sw1i   = (const float*)d_in[4];
    const float* sw2r   = (const float*)d_in[5];
    const float* sw2i   = (const float*)d_in[6];
    const float* cw     = (const float*)d_in[7];
    const float* cb     = (const float*)d_in[8];
    const float* mlp0_w = (const float*)d_in[9];
    const float* mlp0_b = (const float*)d_in[10];
    const float* mlp1_w = (const float*)d_in[11];
    const float* mlp1_b = (const float*)d_in[12];
    float* out = (float*)d_out;

    float* w = (float*)d_ws;
    size_t off = 0;
    float*  h    = w + off; off += HBUF;                    // (B,32,H,W)
    float*  h2   = w + off; off += HBUF;                    // bypass buffer
    float*  T    = w + off; off += (size_t)BB * 32 * 256 * 32; // stage-A out (ld 32)
    float2* HF   = (float2*)(w + off); off += 2 * BB * 32 * 24 * 12;
    float2* OUTF = (float2*)(w + off); off += 2 * BB * 32 * 24 * 12;
    float*  G    = w + off; off += (size_t)BB * 32 * 256 * 24;
    float*  E2f  = w + off; off += 256 * 32;
    float2* E1f  = (float2*)(w + off); off += 2 * 24 * 256;
    float2* EI1  = (float2*)(w + off); off += 2 * 24 * 256;
    float*  Cinv = w + off; off += 24 * 256;
    float*  W1p  = w + off; off += 128 * 16;
    float*  B1p  = w + off; off += 64;
    float*  H1   = w + off; off += (size_t)32768 * 128;     // MLP hidden chunk
    if (ws_size < off * sizeof(float)) return;

    dim3 gthr(32, 4);

    // Coefficient tables
    {
        int n = 256 * 32 + 3 * (24 * 256) + 128 * 16 + 16;
        init_k<<<(n + 255) / 256, 256, 0, stream>>>(E2f, E1f, EI1, Cinv, W1p, B1p,
                                                    mlp1_w, mlp1_b);
    }
    // Lift
    lift_k<<<HBUF / 256, 256, 0, stream>>>(xin, p_w, p_b, h);

    for (int l = 0; l < LL; ++l) {
        // Stage A: T = h @ E2f   (M=32768, K=256(y), N=32 pad)   NT=2 -> 2048 waves
        gemm_wmma_k<2><<<dim3(512, 1, 1), gthr, 0, stream>>>(
            h, E2f, nullptr, nullptr, T,
            2048, 1, 256, 32, 0,
            256LL, 1LL, 0LL, 32LL, 32LL, 1LL, 0LL);

        stageB_k<<<(BB * 32 * 288) / 256, 256, 0, stream>>>(T, E1f, HF);
        mix_k<<<(BB * 32 * 288) / 256, 256, 0, stream>>>(HF, sw1r, sw1i, sw2r, sw2i,
                                                         OUTF, l);
        stageC_k<<<(BB * 32 * 256) / 256, 256, 0, stream>>>(OUTF, EI1, G);

        // Bypass: h2[b,o,xy] = sum_i h[b,i,xy]*cw[l,i,o] + cb[l,o]   batched over b
        gemm_wmma_k<2><<<dim3(1024, 1, BB), gthr, 0, stream>>>(
            h, cw + l * 1024, cb + l * 32, nullptr, h2,
            4096, 1, 32, 32, 1,
            1LL, (long long)HWSZ, (long long)WIDTH * HWSZ,
            32LL,
            1LL, (long long)HWSZ, (long long)WIDTH * HWSZ);

        // Stage D: h = (G @ Cinv) + h2, gelu for l<3   (M=32768, K=24, N=256) NT=4
        gemm_wmma_k<4><<<dim3(2048, 1, 1), gthr, 0, stream>>>(
            G, Cinv, nullptr, h2, h,
            2048, 4, 24, 256, (l < LL - 1) ? 4 : 3,
            24LL, 1LL, 0LL, 256LL, 256LL, 1LL, 0LL);
    }

    // Pointwise MLP in 8 chunks of 32768 rows (each chunk within one batch b)
    for (int c = 0; c < 8; ++c) {
        long long b = c / 2;
        long long xy0 = (long long)(c % 2) * 32768;
        const float* Ain = h + b * (WIDTH * HWSZ) + xy0;
        // MLP0: H1 = gelu(Ain^T(32) @ mlp0_w + b)   M=32768, K=32, N=128, NT=4
        gemm_wmma_k<4><<<dim3(1024, 1, 1), gthr, 0, stream>>>(
            Ain, mlp0_w, mlp0_b, nullptr, H1,
            2048, 2, 32, 128, 2,
            1LL, (long long)HWSZ, 0LL, 128LL, 128LL, 1LL, 0LL);
        // MLP1: out = H1 @ W1p + B1p   M=32768, K=128, N=16 (valid 3)
        float* Cout = out + b * (CC * HWSZ) + xy0;
        gemm_wmma_k<1><<<dim3(512, 1, 1), gthr, 0, stream>>>(
            H1, W1p, B1p, nullptr, Cout,
            2048, 1, 128, 3, 1,
            128LL, 1LL, 0LL, 16LL, 1LL, (long long)HWSZ, 0LL);
    }
}